// GraphSAGEModelV0_68977174774176
// MI455X (gfx1250) — compile-verified
//
#include <hip/hip_runtime.h>

// ---------------------------------------------------------------------------
// GraphSAGE (3-layer, mean aggregation) for MI455X / gfx1250.
//  - Aggregation: wave-per-edge-row scatter with global_atomic_add_f32
//  - Dense part:  fused [mean | x] @ [Wl;Wr] GEMM using V_WMMA_F32_16X16X4_F32
//    (compile-time K/N_OUT, 2 M-subtiles per wave, unrolled K-loop,
//     branch-free epilogue: grid covers N exactly, 100000 % 32 == 0)
// ---------------------------------------------------------------------------

typedef __attribute__((ext_vector_type(2))) float v2f;
typedef __attribute__((ext_vector_type(8))) float v8f;

#define D_IN 128
#define KDIM 256           // fused K = 2 * D_IN  ([mean | x])
#define LDSK 260           // padded LDS row stride: bank-conflict-free, 16B aligned

// ---------------- degree ----------------------------------------------------
__global__ void sage_degree_kernel(const int* __restrict__ dst, float* __restrict__ cnt,
                                   long n_edges) {
    long tid = (long)blockIdx.x * blockDim.x + threadIdx.x;
    if (tid < n_edges) {
        atomicAdd(&cnt[dst[tid]], 1.0f);
    }
}

__global__ void sage_invdeg_kernel(float* __restrict__ cnt, int n_nodes) {
    int tid = blockIdx.x * blockDim.x + threadIdx.x;
    if (tid < n_nodes) {
        cnt[tid] = 1.0f / fmaxf(cnt[tid], 1.0f);   // becomes inv-degree in place
    }
}

// ---------------- scatter-add aggregation -----------------------------------
// One wave (32 lanes) handles one edge's full 128-float feature row:
// lane L moves floats [4L, 4L+4). src/dst loads are wave-uniform (scalarized),
// the gather is one coalesced 512B row read, the scatter is 4 f32 atomics/lane.
__global__ void sage_scatter_kernel(const float* __restrict__ h,
                                    const int* __restrict__ src,
                                    const int* __restrict__ dst,
                                    float* __restrict__ agg,
                                    long n_edges) {
    long tid   = (long)blockIdx.x * blockDim.x + threadIdx.x;
    long edge  = tid >> 5;           // 32 lanes per edge (d=128 -> 32 float4 chunks)
    int  chunk = (int)(tid & 31);
    if (edge >= n_edges) return;

    int s = src[edge];
    int d = dst[edge];

    const float4* row = (const float4*)(h + (size_t)s * D_IN);
    float4 v = row[chunk];

    float* arow = agg + (size_t)d * D_IN + chunk * 4;
    atomicAdd(arow + 0, v.x);
    atomicAdd(arow + 1, v.y);
    atomicAdd(arow + 2, v.z);
    atomicAdd(arow + 3, v.w);
}

// ---------------- pack [Wl; Wr] transposed into Wcat[k][o] ------------------
// Wl, Wr are [n_out][d_in] row-major. Wcat is [2*d_in][n_out]:
//   Wcat[k][o] = (k < d_in) ? Wl[o][k] : Wr[o][k - d_in]
__global__ void sage_wcat_kernel(const float* __restrict__ Wl,
                                 const float* __restrict__ Wr,
                                 float* __restrict__ Wcat,
                                 int d_in, int n_out) {
    int idx = blockIdx.x * blockDim.x + threadIdx.x;
    int K = 2 * d_in;
    if (idx >= K * n_out) return;
    int k = idx / n_out;
    int o = idx % n_out;
    Wcat[idx] = (k < d_in) ? Wl[o * d_in + k] : Wr[o * d_in + (k - d_in)];
}

// ---------------- fused SAGE GEMM via WMMA f32 16x16x4 ----------------------
// Block: M_TILE = 16*M_SUB node rows; one wave per 16-column output subtile,
// each wave accumulates M_SUB independent 16x16 tiles (B-fragment reuse + ILP).
//   out[i] = act( [agg[i]*inv | x[i]] @ Wcat + b )
// NOTE: grid must cover n_nodes exactly (n_nodes % (16*M_SUB) == 0) -- the
// loader and epilogue are branch-free; EXEC stays all-ones throughout.
template <int N_OUT, int M_SUB, bool RELU>
__global__ void __launch_bounds__(32 * (N_OUT / 16))
sage_gemm_kernel(const float* __restrict__ x,     // [N][128] self feats
                 const float* __restrict__ agg,   // [N][128] summed msgs
                 const float* __restrict__ inv,   // [N] 1/max(deg,1)
                 const float* __restrict__ Wcat,  // [256][N_OUT]
                 const float* __restrict__ bias,  // [N_OUT]
                 float* __restrict__ out)         // [N][N_OUT]
{
    extern __shared__ float lds[];                // [M_TILE][LDSK]
    constexpr int NTHREADS = 32 * (N_OUT / 16);
    constexpr int M_TILE   = 16 * M_SUB;

    const int row0 = blockIdx.x * M_TILE;
    const int tid  = threadIdx.x;

    // Stage A tile with float4: cols [0,128) = mean = agg*inv ; [128,256) = x
#pragma unroll
    for (int i = tid; i < M_TILE * (KDIM / 4); i += NTHREADS) {
        int r = i >> 6;                 // KDIM/4 = 64 quads per row
        int k = (i & 63) << 2;
        int node = row0 + r;
        float4 v;
        if (k < D_IN) {
            float s = inv[node];
            v = *(const float4*)(agg + (size_t)node * D_IN + k);
            v.x *= s; v.y *= s; v.z *= s; v.w *= s;
        } else {
            v = *(const float4*)(x + (size_t)node * D_IN + (k - D_IN));
        }
        *(float4*)(&lds[r * LDSK + k]) = v;
    }
    __syncthreads();

    const int lane  = tid & 31;
    const int wave  = tid >> 5;
    const int col   = wave * 16 + (lane & 15);  // output column (B/D lane)
    const int m     = lane & 15;                // A-fragment row for this lane
    const int khalf = (lane >> 4) << 1;         // lanes 0-15 -> K0/K1, 16-31 -> K2/K3

    const v8f zero = {0.f, 0.f, 0.f, 0.f, 0.f, 0.f, 0.f, 0.f};
    v8f c[M_SUB];
#pragma unroll
    for (int mt = 0; mt < M_SUB; ++mt) c[mt] = zero;

#pragma unroll 4
    for (int k0 = 0; k0 < KDIM; k0 += 4) {
        v2f b;
        b.x = Wcat[(k0 + khalf) * N_OUT + col];
        b.y = Wcat[(k0 + khalf + 1) * N_OUT + col];
#pragma unroll
        for (int mt = 0; mt < M_SUB; ++mt) {
            const float* arow = &lds[(mt * 16 + m) * LDSK + k0 + khalf];
            v2f a;
            a.x = arow[0];
            a.y = arow[1];
            // D = A(16x4,f32) * B(4x16,f32) + C ; EXEC is all-ones here.
            c[mt] = __builtin_amdgcn_wmma_f32_16x16x4_f32(
                        false, a, false, b, (short)0, c[mt], false, false);
        }
    }

    // Branch-free epilogue. C/D layout: VGPR r holds M=r (lanes 0-15) / M=r+8
    // (lanes 16-31); row stride folds into the store's immediate offset.
    const float bv   = bias[col];
    const int   mofs = (lane >> 4) << 3;
#pragma unroll
    for (int mt = 0; mt < M_SUB; ++mt) {
        float* obase = out + (size_t)(row0 + mt * 16 + mofs) * N_OUT + col;
#pragma unroll
        for (int r = 0; r < 8; ++r) {
            float v = c[mt][r] + bv;
            if (RELU) v = fmaxf(v, 0.0f);
            obase[r * N_OUT] = v;
        }
    }
}

// ---------------------------------------------------------------------------
extern "C" void kernel_launch(void* const* d_in, const int* in_sizes, int n_in,
                              void* d_out, int out_size, void* d_ws, size_t ws_size,
                              hipStream_t stream) {
    const float* x   = (const float*)d_in[0];
    const int*   ei  = (const int*)d_in[1];
    const float* Wl0 = (const float*)d_in[2];
    const float* Wr0 = (const float*)d_in[3];
    const float* b0  = (const float*)d_in[4];
    const float* Wl1 = (const float*)d_in[5];
    const float* Wr1 = (const float*)d_in[6];
    const float* b1  = (const float*)d_in[7];
    const float* Wl2 = (const float*)d_in[8];
    const float* Wr2 = (const float*)d_in[9];
    const float* b2  = (const float*)d_in[10];

    const int  N = in_sizes[0] / D_IN;      // 100000 (multiple of 32)
    const long E = in_sizes[1] / 2;         // 1600000
    const int* src = ei;
    const int* dst = ei + E;

    // Workspace layout (floats): agg[N*128] | h[N*128] | inv[N] | Wcat[256*128]
    float* ws   = (float*)d_ws;
    float* agg  = ws;
    float* h    = agg + (size_t)N * D_IN;
    float* inv  = h   + (size_t)N * D_IN;
    float* wcat = inv + (size_t)N;

    // ---- degree / inverse degree (shared by all layers) ----
    hipMemsetAsync(inv, 0, (size_t)N * sizeof(float), stream);
    sage_degree_kernel<<<(int)((E + 255) / 256), 256, 0, stream>>>(dst, inv, E);
    sage_invdeg_kernel<<<(N + 255) / 256, 256, 0, stream>>>(inv, N);

    const long scatThreads = E * 32;
    const int  scatBlocks  = (int)((scatThreads + 255) / 256);

    constexpr int M_SUB   = 2;              // 32-row tiles; 100000 % 32 == 0
    const int  gemmBlocks = N / (16 * M_SUB);                      // 3125 exact
    const size_t shmem    = (size_t)(16 * M_SUB) * LDSK * sizeof(float); // 33.3 KB

    // ---- layer 0: x -> h (relu) ----
    hipMemsetAsync(agg, 0, (size_t)N * D_IN * sizeof(float), stream);
    sage_scatter_kernel<<<scatBlocks, 256, 0, stream>>>(x, src, dst, agg, E);
    sage_wcat_kernel<<<(256 * 128 + 255) / 256, 256, 0, stream>>>(Wl0, Wr0, wcat, D_IN, 128);
    sage_gemm_kernel<128, M_SUB, true><<<gemmBlocks, 256, shmem, stream>>>(
        x, agg, inv, wcat, b0, h);

    // ---- layer 1: h -> h (relu, in place; each block stages its rows in LDS) ----
    hipMemsetAsync(agg, 0, (size_t)N * D_IN * sizeof(float), stream);
    sage_scatter_kernel<<<scatBlocks, 256, 0, stream>>>(h, src, dst, agg, E);
    sage_wcat_kernel<<<(256 * 128 + 255) / 256, 256, 0, stream>>>(Wl1, Wr1, wcat, D_IN, 128);
    sage_gemm_kernel<128, M_SUB, true><<<gemmBlocks, 256, shmem, stream>>>(
        h, agg, inv, wcat, b1, h);

    // ---- layer 2: h -> out (n_out = 64, no activation) ----
    hipMemsetAsync(agg, 0, (size_t)N * D_IN * sizeof(float), stream);
    sage_scatter_kernel<<<scatBlocks, 256, 0, stream>>>(h, src, dst, agg, E);
    sage_wcat_kernel<<<(256 * 64 + 255) / 256, 256, 0, stream>>>(Wl2, Wr2, wcat, D_IN, 64);
    sage_gemm_kernel<64, M_SUB, false><<<gemmBlocks, 128, shmem, stream>>>(
        h, agg, inv, wcat, b2, (float*)d_out);
}